// KMeans_69595650064679
// MI455X (gfx1250) — compile-verified
//
#include <hip/hip_runtime.h>

// KMeans / Lloyd's algorithm for MI455X (gfx1250, wave32, WMMA).
//   N=500000 points, M=64 dims, K=256 clusters, 9 Lloyd steps.
// GEMM core: V_WMMA_F32_16X16X4_F32 (exact f32), K-loop of 16 per 16x16 tile.
// Fused assignment + segment-sum using 320KB WGP LDS (cent 64KB + sums 64KB).

#define KM_K      256
#define KM_M      64
#define KM_N      500000
#define KM_ITERS  9
#define KM_KM     (KM_K * KM_M)   // 16384

typedef float v2f __attribute__((ext_vector_type(2)));
typedef float v8f __attribute__((ext_vector_type(8)));

// ---------------------------------------------------------------------------
// Zero the global accumulators.
__global__ __launch_bounds__(256) void kmeans_zero(float* __restrict__ g_sums,
                                                   float* __restrict__ g_counts) {
    int i = blockIdx.x * blockDim.x + threadIdx.x;
    if (i < KM_KM) g_sums[i] = 0.0f;
    if (i < KM_K)  g_counts[i] = 0.0f;
}

// ---------------------------------------------------------------------------
// Fused: distances via WMMA f32 -> argmin -> LDS segment-sum -> global flush.
// Block = 256 threads = 8 waves. Each wave processes 16-point groups.
__global__ __launch_bounds__(256) void kmeans_assign_accum(
    const float* __restrict__ data,      // [N][64]
    const float* __restrict__ cent,      // [256][64]
    float* __restrict__ g_sums,          // [256][64]
    float* __restrict__ g_counts,        // [256]
    float* __restrict__ out_assign)      // [N] (as float)
{
    extern __shared__ float smem[];
    float* lds_cent = smem;              // 16384 f32 (64KB)
    float* lds_sums = smem + KM_KM;      // 16384 f32 (64KB)
    float* lds_c2   = smem + 2 * KM_KM;  // 256 f32
    float* lds_cnt  = lds_c2 + KM_K;     // 256 f32

    const int tid = threadIdx.x;

    // Stage centroids to LDS, zero LDS accumulators.
    for (int i = tid; i < KM_KM; i += 256) {
        lds_cent[i] = cent[i];
        lds_sums[i] = 0.0f;
    }
    lds_cnt[tid] = 0.0f;
    __syncthreads();

    // c2[k] = ||cent_k||^2  (one cluster per thread, from LDS)
    {
        const float* row = lds_cent + tid * KM_M;
        float s = 0.0f;
        #pragma unroll
        for (int m = 0; m < KM_M; ++m) { float v = row[m]; s = __builtin_fmaf(v, v, s); }
        lds_c2[tid] = s;
    }
    __syncthreads();

    const int lane = tid & 31;
    const int hv   = lane >> 4;   // half-wave: 0 or 1
    const int r    = lane & 15;   // row within half

    const int gwave   = blockIdx.x * (blockDim.x >> 5) + (tid >> 5);
    const int nwaves  = gridDim.x * (blockDim.x >> 5);
    const int ngroups = KM_N / 16;  // 31250 groups of 16 points

    for (int g = gwave; g < ngroups; g += nwaves) {
        const int p0 = g * 16;

        // A fragments: 16x4 f32 tiles, 16 k-steps covering M=64.
        // ISA layout: lane<16 VGPR{0,1} = A[M=lane][k0,k0+1]; lane>=16 = [k0+2,k0+3].
        v2f a[16];
        const float* arow = data + (size_t)(p0 + r) * KM_M + hv * 2;
        #pragma unroll
        for (int ks = 0; ks < 16; ++ks)
            a[ks] = *(const v2f*)(arow + ks * 4);

        float bests[8];
        int   bestc[8];
        #pragma unroll
        for (int j = 0; j < 8; ++j) { bests[j] = 3.4e38f; bestc[j] = 0; }

        // Sweep all 256 clusters in 16-wide tiles; A reused across all tiles.
        for (int c0 = 0; c0 < KM_K; c0 += 16) {
            v8f acc = {0.f, 0.f, 0.f, 0.f, 0.f, 0.f, 0.f, 0.f};
            // B fragments from LDS (B = cent^T, same per-lane addressing as A).
            const float* brow = lds_cent + (size_t)(c0 + r) * KM_M + hv * 2;
            #pragma unroll
            for (int ks = 0; ks < 16; ++ks) {
                v2f b = *(const v2f*)(brow + ks * 4);
                acc = __builtin_amdgcn_wmma_f32_16x16x4_f32(
                        /*neg_a=*/false, a[ks], /*neg_b=*/false, b,
                        /*c_mod=*/(short)0, acc, /*reuse_a=*/false, /*reuse_b=*/false);
            }
            // C layout: lane holds cluster column n = c0 + r; VGPR j -> point m = j + 8*hv.
            // score = ||c||^2 - 2*x.c  (||x||^2 constant per point; sqrt monotone -> both dropped)
            const float c2v = lds_c2[c0 + r];
            #pragma unroll
            for (int j = 0; j < 8; ++j) {
                float s = __builtin_fmaf(-2.0f, acc[j], c2v);
                if (s < bests[j]) { bests[j] = s; bestc[j] = c0 + r; }
            }
        }

        // Argmin across the 16 lanes of each half-wave (butterfly; ties -> lowest cluster).
        #pragma unroll
        for (int j = 0; j < 8; ++j) {
            float s = bests[j]; int c = bestc[j];
            #pragma unroll
            for (int off = 1; off < 16; off <<= 1) {
                float so = __shfl_xor(s, off, 16);
                int   co = __shfl_xor(c, off, 16);
                if (so < s || (so == s && co < c)) { s = so; c = co; }
            }
            bests[j] = s; bestc[j] = c;
        }

        // Accumulate: 16 lanes of each half cooperatively add point p into LDS sums.
        #pragma unroll
        for (int j = 0; j < 8; ++j) {
            const int p = p0 + hv * 8 + j;
            const int c = bestc[j];                     // uniform within half-wave
            float4 d = *(const float4*)(data + (size_t)p * KM_M + r * 4);
            float* dst = lds_sums + (size_t)c * KM_M + r * 4;
            atomicAdd(dst + 0, d.x);
            atomicAdd(dst + 1, d.y);
            atomicAdd(dst + 2, d.z);
            atomicAdd(dst + 3, d.w);
            if (r == 0) {
                atomicAdd(&lds_cnt[c], 1.0f);
                out_assign[p] = (float)c;               // last iteration's write survives
            }
        }
    }

    __syncthreads();
    // Flush block-local partials to global accumulators.
    for (int i = tid; i < KM_KM; i += 256)
        atomicAdd(&g_sums[i], lds_sums[i]);
    atomicAdd(&g_counts[tid], lds_cnt[tid]);
}

// ---------------------------------------------------------------------------
// new_cent = counts>0 ? sums/counts : 0. Optionally mirror into d_out (last iter).
__global__ __launch_bounds__(256) void kmeans_finalize(
    const float* __restrict__ g_sums, const float* __restrict__ g_counts,
    float* __restrict__ cent_next, float* __restrict__ out_or_null)
{
    int i = blockIdx.x * blockDim.x + threadIdx.x;
    if (i >= KM_KM) return;
    float cnt = g_counts[i >> 6];  // i / KM_M
    float v = (cnt > 0.5f) ? g_sums[i] / cnt : 0.0f;
    cent_next[i] = v;
    if (out_or_null) out_or_null[i] = v;
}

// ---------------------------------------------------------------------------
extern "C" void kernel_launch(void* const* d_in, const int* in_sizes, int n_in,
                              void* d_out, int out_size, void* d_ws, size_t ws_size,
                              hipStream_t stream) {
    const float* data  = (const float*)d_in[0];   // [500000*64] f32
    const float* cinit = (const float*)d_in[1];   // [256*64] f32

    float* out_cent   = (float*)d_out;            // first 16384 floats
    float* out_assign = out_cent + KM_KM;         // next 500000 floats

    float* ws       = (float*)d_ws;
    float* centA    = ws;                  // 16384
    float* centB    = ws + KM_KM;          // 16384
    float* g_sums   = ws + 2 * KM_KM;      // 16384
    float* g_counts = ws + 3 * KM_KM;      // 256

    hipMemcpyAsync(centA, cinit, KM_KM * sizeof(float),
                   hipMemcpyDeviceToDevice, stream);

    const size_t lds_bytes = (2 * KM_KM + 2 * KM_K) * sizeof(float); // 130KB of 320KB/WGP
    float* cur = centA;
    float* nxt = centB;

    for (int t = 0; t < KM_ITERS; ++t) {
        kmeans_zero<<<(KM_KM + 255) / 256, 256, 0, stream>>>(g_sums, g_counts);
        kmeans_assign_accum<<<256, 256, lds_bytes, stream>>>(
            data, cur, g_sums, g_counts, out_assign);
        kmeans_finalize<<<KM_KM / 256, 256, 0, stream>>>(
            g_sums, g_counts, nxt, (t == KM_ITERS - 1) ? out_cent : nullptr);
        float* tmp = cur; cur = nxt; nxt = tmp;
    }
}